// XORNet_37615323578711
// MI455X (gfx1250) — compile-verified
//
#include <hip/hip_runtime.h>
#include <stdint.h>

// ---------------------------------------------------------------------------
// XORNet fused forward for MI455X (gfx1250).
// Roofline: 100 MB traffic @ 23.3 TB/s ~ 4.3us floor; compute side is
// trans-pipe-limited (17 sigmoids/row -> one v_tanh_f32 each).  FMA chains
// are packed into v_pk_fma_f32 pairs (two independent rows per float4) so
// instruction issue stays below the quarter-rate trans pipe.
// Input staged via the CDNA5 async global->LDS path (ASYNCcnt),双-buffered.
// ---------------------------------------------------------------------------

typedef __attribute__((ext_vector_type(4))) float float4v;
typedef __attribute__((ext_vector_type(2))) float float2v;
typedef __attribute__((ext_vector_type(4))) int   int4v;

// Address-space-qualified types for the async-copy builtin (params are
// v4i addrspace(1)* and v4i addrspace(3)*).
typedef __attribute__((address_space(1))) int4v int4_global;
typedef __attribute__((address_space(3))) int4v int4_lds;

// ---- hardware tanh (gfx1250 TRANS op) -------------------------------------
__device__ __forceinline__ float htanh(float x) {
#if __has_builtin(__builtin_amdgcn_tanhf)
  return __builtin_amdgcn_tanhf(x);
#elif __has_builtin(__builtin_amdgcn_tanh_f32)
  return __builtin_amdgcn_tanh_f32(x);
#else
  float r;
  asm("v_tanh_f32 %0, %1" : "=v"(r) : "v"(x));
  return r;
#endif
}

__device__ __forceinline__ float2v splat2(float s) {
  float2v v = {s, s};
  return v;
}
// Packed fp32 fma: lowers to v_pk_fma_f32 on gfx1250.
__device__ __forceinline__ float2v pkfma(float2v a, float2v b, float2v c) {
  return __builtin_elementwise_fma(a, b, c);
}
__device__ __forceinline__ float2v tanh2(float2v y) {
  float2v t = {htanh(y.x), htanh(y.y)};
  return t;
}

// ---- async global -> LDS, 16B per lane (tracked with ASYNCcnt) ------------
#if __has_builtin(__builtin_amdgcn_global_load_async_to_lds_b128)
#define ASYNC_B128(gsrc, ldst)                                              \
  __builtin_amdgcn_global_load_async_to_lds_b128(                           \
      (int4_global*)(uintptr_t)(gsrc),                                      \
      (int4_lds*)(unsigned)(uintptr_t)(ldst), 0, 0)
#else
#define ASYNC_B128(gsrc, ldst)                                              \
  asm volatile("global_load_async_to_lds_b128 %0, %1, off" ::               \
                   "v"((unsigned)(uintptr_t)(ldst)),                        \
                   "v"((const void*)(gsrc))                                 \
               : "memory")
#endif

#if __has_builtin(__builtin_amdgcn_s_wait_asynccnt)
#define WAIT_ASYNC(n)                        \
  do {                                       \
    __builtin_amdgcn_s_wait_asynccnt(n);     \
    asm volatile("" ::: "memory");           \
  } while (0)
#else
#define WAIT_ASYNC(n) asm volatile("s_wait_asynccnt %0" ::"n"(n) : "memory")
#endif

// Each block: 256 threads, 2 stages x 2 slots x 1 float4 (=2 rows) per thread
// => 2048 rows per block.  LDS = 2*2*256*16B = 16 KB.
__global__ __launch_bounds__(256) void xornet_fused(
    const float* __restrict__ x,   // [B,2]
    const float* __restrict__ Ws,  // [8,2,2]  (y_o = sum_i h_i * W[o][i])
    const float* __restrict__ bs,  // [8,2]
    const float* __restrict__ W8,  // [1,2]
    const float* __restrict__ b8,  // [1]
    float* __restrict__ out,       // [B,1]
    int n_f4)                      // B/2 float4 chunks of x
{
  __shared__ float lds[2 * 2 * 256 * 4];

  const int t = threadIdx.x;
  const int base = blockIdx.x * 1024;  // first float4 index for this block

  // Pre-halved weights: sigmoid(y) = 0.5*tanh(y/2) + 0.5, with y/2 produced
  // directly by the FMA chain.  Constant indices -> scalar (s_load) path.
  float w00[8], w01[8], w10[8], w11[8], c0[8], c1[8];
#pragma unroll
  for (int l = 0; l < 8; ++l) {
    w00[l] = 0.5f * Ws[4 * l + 0];
    w01[l] = 0.5f * Ws[4 * l + 1];
    w10[l] = 0.5f * Ws[4 * l + 2];
    w11[l] = 0.5f * Ws[4 * l + 3];
    c0[l]  = 0.5f * bs[2 * l + 0];
    c1[l]  = 0.5f * bs[2 * l + 1];
  }
  const float v0 = 0.5f * W8[0];
  const float v1 = 0.5f * W8[1];
  const float vb = 0.5f * b8[0];
  const float2v half2 = {0.5f, 0.5f};

  // Kick off both stages' async copies up front (loads complete in order,
  // so waiting asynccnt<=2 releases stage 0, <=0 releases stage 1).
#pragma unroll
  for (int u = 0; u < 2; ++u) {
    const int idx = base + u * 256 + t;
    if (idx < n_f4)
      ASYNC_B128(x + 4ll * idx, &lds[(u * 256 + t) * 4]);
  }
#pragma unroll
  for (int u = 0; u < 2; ++u) {
    const int idx = base + 512 + u * 256 + t;
    if (idx < n_f4)
      ASYNC_B128(x + 4ll * idx, &lds[((2 + u) * 256 + t) * 4]);
  }

#pragma unroll
  for (int s = 0; s < 2; ++s) {
    if (s == 0) { WAIT_ASYNC(2); } else { WAIT_ASYNC(0); }
#pragma unroll
    for (int u = 0; u < 2; ++u) {
      const int idx = base + s * 512 + u * 256 + t;
      if (idx >= n_f4) continue;

      // One float4 = two independent rows -> pack them as the two lanes of
      // v_pk_fma_f32 operands.  h0 = (rowA.h0, rowB.h0), h1 = (rowA.h1, ...).
      float4v d = *(float4v*)&lds[((s * 2 + u) * 256 + t) * 4];
      float2v h0 = {d.x, d.z};
      float2v h1 = {d.y, d.w};

#pragma unroll
      for (int l = 0; l < 8; ++l) {
        // yh = 0.5*(W h + b); h' = 0.5*tanh(yh) + 0.5 == sigmoid(W h + b)
        float2v y0 = pkfma(splat2(w00[l]), h0, pkfma(splat2(w01[l]), h1, splat2(c0[l])));
        float2v y1 = pkfma(splat2(w10[l]), h0, pkfma(splat2(w11[l]), h1, splat2(c1[l])));
        h0 = pkfma(half2, tanh2(y0), half2);
        h1 = pkfma(half2, tanh2(y1), half2);
      }
      float2v y = pkfma(splat2(v0), h0, pkfma(splat2(v1), h1, splat2(vb)));
      float2v r = pkfma(half2, tanh2(y), half2);
      *(float2v*)(out + 2ll * idx) = r;  // coalesced 8B store
    }
  }
}

extern "C" void kernel_launch(void* const* d_in, const int* in_sizes, int n_in,
                              void* d_out, int out_size, void* d_ws,
                              size_t ws_size, hipStream_t stream) {
  (void)n_in; (void)out_size; (void)d_ws; (void)ws_size;
  const float* x  = (const float*)d_in[0];  // [B,2]
  const float* Ws = (const float*)d_in[1];  // [8,2,2]
  const float* bs = (const float*)d_in[2];  // [8,2]
  const float* W8 = (const float*)d_in[3];  // [1,2]
  const float* b8 = (const float*)d_in[4];  // [1]
  float* out = (float*)d_out;

  const int n_rows = in_sizes[0] / 2;
  const int n_f4 = n_rows / 2;                  // B = 8388608 -> 4194304
  int blocks = (n_f4 + 1023) / 1024;            // 4096 blocks
  if (blocks < 1) blocks = 1;
  xornet_fused<<<blocks, 256, 0, stream>>>(x, Ws, bs, W8, b8, out, n_f4);
}